// HeteroSAGE_12077448036842
// MI455X (gfx1250) — compile-verified
//
#include <hip/hip_runtime.h>

#define HD   128
#define NU   100000
#define NI   100000
#define NE   600000
#define H2   (HD * HD)

typedef __attribute__((ext_vector_type(2))) float v2f;
typedef __attribute__((ext_vector_type(8))) float v8f;

// ---------------------------------------------------------------------------
// GEMM: C[M,128] = (acc ? C : 0) + A0@B0 [+ A1@B1] (+bias0+bias1) (relu)
// A row-major [M,K] with optional per-row scale (inverse degree for the
// mean-neighbor term). B row-major [K,128]. One 256-thread block = 8 waves
// computes a 16-row x 128-col tile; wave w owns cols [16w, 16w+16).
// Uses V_WMMA_F32_16X16X4_F32 (wave32). No early exits: EXEC stays all-ones.
// ---------------------------------------------------------------------------
__global__ __launch_bounds__(256)
void hsage_gemm2(const float* __restrict__ A0, int lda0, int K0,
                 const float* __restrict__ B0,
                 const float* __restrict__ scale0,
                 const float* __restrict__ A1, int lda1, int K1,
                 const float* __restrict__ B1,
                 const float* __restrict__ scale1,
                 const float* __restrict__ bias0,
                 const float* __restrict__ bias1,
                 float* __restrict__ C,
                 int accFlag, int reluFlag)
{
    const int lane = threadIdx.x & 31;
    const int wave = threadIdx.x >> 5;      // 0..7 -> 16-col tile
    const int r    = lane & 15;
    const int half = lane >> 4;             // selects K pair {0,1} vs {2,3}
    const int row0 = blockIdx.x * 16;
    const int col  = wave * 16 + r;

    v8f acc = {0.f, 0.f, 0.f, 0.f, 0.f, 0.f, 0.f, 0.f};
    if (accFlag) {
        #pragma unroll
        for (int i = 0; i < 8; ++i)
            acc[i] = C[(size_t)(row0 + half * 8 + i) * HD + col];
    }

    // ---- term 0: A0 @ B0 ----
    {
        const float* arow = A0 + (size_t)(row0 + r) * lda0;
        const float  sc   = scale0 ? scale0[row0 + r] : 1.0f;
        for (int k = 0; k < K0; k += 4) {
            const float2 av = *(const float2*)(arow + k + 2 * half);
            v2f a, b;
            a.x = av.x * sc;
            a.y = av.y * sc;
            b.x = B0[(size_t)(k + 2 * half) * HD + col];
            b.y = B0[(size_t)(k + 2 * half + 1) * HD + col];
            acc = __builtin_amdgcn_wmma_f32_16x16x4_f32(
                false, a, false, b, (short)0, acc, false, false);
        }
    }
    // ---- term 1: A1 @ B1 (optional) ----
    if (A1 != nullptr) {
        const float* arow = A1 + (size_t)(row0 + r) * lda1;
        const float  sc   = scale1 ? scale1[row0 + r] : 1.0f;
        for (int k = 0; k < K1; k += 4) {
            const float2 av = *(const float2*)(arow + k + 2 * half);
            v2f a, b;
            a.x = av.x * sc;
            a.y = av.y * sc;
            b.x = B1[(size_t)(k + 2 * half) * HD + col];
            b.y = B1[(size_t)(k + 2 * half + 1) * HD + col];
            acc = __builtin_amdgcn_wmma_f32_16x16x4_f32(
                false, a, false, b, (short)0, acc, false, false);
        }
    }

    float bsum = 0.f;
    if (bias0) bsum += bias0[col];
    if (bias1) bsum += bias1[col];

    #pragma unroll
    for (int i = 0; i < 8; ++i) {
        float v = acc[i] + bsum;
        if (reluFlag) v = fmaxf(v, 0.f);
        C[(size_t)(row0 + half * 8 + i) * HD + col] = v;
    }
}

// ---------------------------------------------------------------------------
// Zero a buffer (float4 granularity; all sizes here are multiples of 4).
// ---------------------------------------------------------------------------
__global__ __launch_bounds__(256)
void hsage_zero4(float4* __restrict__ p, int n4)
{
    int i = blockIdx.x * blockDim.x + threadIdx.x;
    if (i < n4) p[i] = make_float4(0.f, 0.f, 0.f, 0.f);
}

// ---------------------------------------------------------------------------
// Edge scatter-sum: one wave per edge, lane handles 4 floats of the 128-dim
// feature. Lane 0 accumulates the destination in-degree.
// ---------------------------------------------------------------------------
__global__ __launch_bounds__(256)
void hsage_scatter(const float* __restrict__ hsrc,
                   const int* __restrict__ src,
                   const int* __restrict__ dst,
                   float* __restrict__ m,
                   float* __restrict__ deg, int E)
{
    int tid  = blockIdx.x * blockDim.x + threadIdx.x;
    int e    = tid >> 5;
    int lane = tid & 31;
    if (e >= E) return;
    int s = src[e];
    int d = dst[e];
    const float4 v = *(const float4*)(hsrc + (size_t)s * HD + lane * 4);
    float* o = m + (size_t)d * HD + lane * 4;
    atomicAdd(o + 0, v.x);
    atomicAdd(o + 1, v.y);
    atomicAdd(o + 2, v.z);
    atomicAdd(o + 3, v.w);
    if (lane == 0) atomicAdd(deg + d, 1.0f);
}

// deg[i] = 1 / max(deg[i], 1)  -> per-row scale consumed by the GEMM A-load
__global__ __launch_bounds__(256)
void hsage_recip(float* __restrict__ deg, int n)
{
    int i = blockIdx.x * blockDim.x + threadIdx.x;
    if (i < n) deg[i] = 1.0f / fmaxf(deg[i], 1.0f);
}

// ---------------------------------------------------------------------------
// Row-wise L2 normalize: one wave per row, lane holds 4 floats, xor-shuffle
// reduction across the 32 lanes (wave32).
// ---------------------------------------------------------------------------
__global__ __launch_bounds__(256)
void hsage_l2norm(const float* __restrict__ h, float* __restrict__ out, int n)
{
    int tid  = blockIdx.x * blockDim.x + threadIdx.x;
    int row  = tid >> 5;
    int lane = tid & 31;
    if (row >= n) return;
    const float4 v = *(const float4*)(h + (size_t)row * HD + lane * 4);
    float ss = v.x * v.x + v.y * v.y + v.z * v.z + v.w * v.w;
    #pragma unroll
    for (int d = 16; d >= 1; d >>= 1)
        ss += __shfl_xor(ss, d, 32);
    float inv = 1.0f / fmaxf(sqrtf(ss), 1e-12f);
    float4 o = make_float4(v.x * inv, v.y * inv, v.z * inv, v.w * inv);
    *(float4*)(out + (size_t)row * HD + lane * 4) = o;
}

// ---------------------------------------------------------------------------
extern "C" void kernel_launch(void* const* d_in, const int* in_sizes, int n_in,
                              void* d_out, int out_size, void* d_ws, size_t ws_size,
                              hipStream_t stream)
{
    const float* x_user = (const float*)d_in[0];
    const float* x_item = (const float*)d_in[1];
    const float* Wp_u   = (const float*)d_in[2];
    const float* bp_u   = (const float*)d_in[3];
    const float* Wp_i   = (const float*)d_in[4];
    const float* bp_i   = (const float*)d_in[5];
    const float* Ws     = (const float*)d_in[6];   // [2,3,128,128]
    const float* Wn     = (const float*)d_in[7];   // [2,3,128,128]
    const float* Wb     = (const float*)d_in[8];   // [2,3,128]
    const int* e_uu_src = (const int*)d_in[9];
    const int* e_uu_dst = (const int*)d_in[10];
    const int* e_ui_src = (const int*)d_in[11];
    const int* e_ui_dst = (const int*)d_in[12];
    const int* e_iu_src = (const int*)d_in[13];
    const int* e_iu_dst = (const int*)d_in[14];

    const size_t NH = (size_t)NU * HD;     // == NI*HD

    // ws: 4 ping-pong feature buffers (h_u, h_i, tmp_u, tmp_i) = ~205 MB
    float* hu = (float*)d_ws;
    float* hi = hu + NH;
    float* tu = hi + NH;
    float* ti = tu + NH;

    // d_out (2*NH floats) is free until the final L2-norm: use it as the
    // mean-aggregation scratch (m: NH floats) + degree vector (NU floats).
    float* outp = (float*)d_out;
    float* m    = outp;
    float* deg  = outp + NH;

    const dim3 blk(256);
    const int gemmGrid = NU / 16;                      // 6250, exact
    const int scatGrid = (NE * 32) / 256;              // 75000, exact
    const int normGrid = (NU * 32) / 256;              // 12500, exact

    auto gemm = [&](const float* A0, int lda0, int K0, const float* B0, const float* s0,
                    const float* A1, int lda1, int K1, const float* B1, const float* s1,
                    const float* b0, const float* b1, float* C, int acc, int relu) {
        hsage_gemm2<<<gemmGrid, blk, 0, stream>>>(A0, lda0, K0, B0, s0,
                                                  A1, lda1, K1, B1, s1,
                                                  b0, b1, C, acc, relu);
    };
    auto meanagg = [&](const float* hsrc, const int* src, const int* dst, int ndst) {
        int n4m = ndst * (HD / 4);
        hsage_zero4<<<(n4m + 255) / 256, blk, 0, stream>>>((float4*)m, n4m);
        hsage_zero4<<<(ndst / 4 + 255) / 256, blk, 0, stream>>>((float4*)deg, ndst / 4);
        hsage_scatter<<<scatGrid, blk, 0, stream>>>(hsrc, src, dst, m, deg, NE);
        hsage_recip<<<(ndst + 255) / 256, blk, 0, stream>>>(deg, ndst);
    };

    // ---- input projections: h = relu(x @ Wp + bp) ----
    gemm(x_user, 256, 256, Wp_u, nullptr, nullptr, 0, 0, nullptr, nullptr,
         bp_u, nullptr, hu, 0, 1);
    gemm(x_item, 64, 64, Wp_i, nullptr, nullptr, 0, 0, nullptr, nullptr,
         bp_i, nullptr, hi, 0, 1);

    // ---- 2 SAGE layers, ping-pong between (hu,hi) and (tu,ti) ----
    float* cu = hu; float* ci = hi;
    float* nu_ = tu; float* ni_ = ti;
    for (int l = 0; l < 2; ++l) {
        const float* Ws_l = Ws + (size_t)l * 3 * H2;
        const float* Wn_l = Wn + (size_t)l * 3 * H2;
        const float* Wb_l = Wb + (size_t)l * 3 * HD;

        // relation uu (r=0): u -> u
        meanagg(cu, e_uu_src, e_uu_dst, NU);
        gemm(cu, HD, HD, Ws_l + 0 * H2, nullptr,
             m, HD, HD, Wn_l + 0 * H2, deg,
             nullptr, nullptr, nu_, /*acc=*/0, /*relu=*/0);

        // relation iu (r=2): i -> u ; accumulate, add both biases, relu
        meanagg(ci, e_iu_src, e_iu_dst, NU);
        gemm(cu, HD, HD, Ws_l + 2 * H2, nullptr,
             m, HD, HD, Wn_l + 2 * H2, deg,
             Wb_l + 0 * HD, Wb_l + 2 * HD, nu_, /*acc=*/1, /*relu=*/1);

        // relation ui (r=1): u -> i
        meanagg(cu, e_ui_src, e_ui_dst, NI);
        gemm(ci, HD, HD, Ws_l + 1 * H2, nullptr,
             m, HD, HD, Wn_l + 1 * H2, deg,
             Wb_l + 1 * HD, nullptr, ni_, /*acc=*/0, /*relu=*/1);

        // swap ping-pong
        float* t;
        t = cu; cu = nu_; nu_ = t;
        t = ci; ci = ni_; ni_ = t;
    }

    // ---- final row-wise L2 norm -> d_out (u block then i block) ----
    hsage_l2norm<<<normGrid, blk, 0, stream>>>(cu, outp, NU);
    hsage_l2norm<<<normGrid, blk, 0, stream>>>(ci, outp + NH, NI);
}